// FeatureVectorGraft_6914897347048
// MI455X (gfx1250) — compile-verified
//
#include <hip/hip_runtime.h>

#define B_ 4
#define S_ 4096
#define DM 4096
#define DF 512
#define TARGET_SNR_ 0.3f
#define LN_EPS_ 1e-5f
#define NORM_EPS_ 1e-12f

typedef __attribute__((ext_vector_type(2))) float v2f;
typedef __attribute__((ext_vector_type(4))) float v4f;
typedef __attribute__((ext_vector_type(8))) float v8f;

// Deterministic block-wide sum (wave32 shuffle tree + LDS across waves).
__device__ __forceinline__ float block_reduce_sum(float v, float* sdata) {
#pragma unroll
  for (int off = 16; off >= 1; off >>= 1) v += __shfl_xor(v, off, 32);
  int wid = threadIdx.x >> 5;
  int lane = threadIdx.x & 31;
  if (lane == 0) sdata[wid] = v;
  __syncthreads();
  if (wid == 0) {
    int nw = blockDim.x >> 5;
    float r = (lane < nw) ? sdata[lane] : 0.f;
#pragma unroll
    for (int off = 4; off >= 1; off >>= 1) r += __shfl_xor(r, off, 32);
    if (lane == 0) sdata[0] = r;
  }
  __syncthreads();
  float r = sdata[0];
  __syncthreads();
  return r;
}

// Kernel 1: per-batch last-token index, LayerNorm(h), host RMS -> magnitude.
__global__ void __launch_bounds__(256) prep_kernel(
    const float* __restrict__ x, const float* __restrict__ fac,
    const int* __restrict__ mask, const float* __restrict__ gamma,
    const float* __restrict__ beta, float* __restrict__ h,
    float* __restrict__ magnitude, int* __restrict__ last_idx) {
  __shared__ float sdata[32];
  const int b = blockIdx.x;
  const int tid = threadIdx.x;

  // last = clip(sum(mask), 1) - 1   (sum <= 4096, exact in f32)
  float msum = 0.f;
  for (int i = tid; i < S_; i += 256) msum += (float)mask[b * S_ + i];
  float tot = block_reduce_sum(msum, sdata);
  int last = (int)(tot + 0.5f);
  last = (last < 1) ? 1 : last;
  last -= 1;

  // LayerNorm over 512 features (each thread owns 2 elements; two-pass)
  float f0 = fac[b * DF + tid];
  float f1 = fac[b * DF + 256 + tid];
  float mean = block_reduce_sum(f0 + f1, sdata) * (1.f / (float)DF);
  float d0 = f0 - mean, d1 = f1 - mean;
  float var = block_reduce_sum(d0 * d0 + d1 * d1, sdata) * (1.f / (float)DF);
  float rstd = rsqrtf(var + LN_EPS_);
  h[b * DF + tid]       = d0 * rstd * gamma[tid] + beta[tid];
  h[b * DF + 256 + tid] = d1 * rstd * gamma[256 + tid] + beta[256 + tid];

  // RMS of x[b, last, :]
  const float* xr = x + ((size_t)b * S_ + (size_t)last) * DM;
  float ss = 0.f;
  for (int i = tid; i < DM; i += 256) { float v = xr[i]; ss += v * v; }
  float s2 = block_reduce_sum(ss, sdata);
  if (tid == 0) {
    magnitude[b] = sqrtf(s2 * (1.f / (float)DM)) * TARGET_SNR_;
    last_idx[b] = last;
  }
}

// Kernel 2: proj[m,n] = sum_k h[m,k]*W[n,k] + bias[n] via V_WMMA_F32_16X16X4_F32.
// One wave per 16-column tile; K=512 chained as 128 WMMAs on the f32 matrix pipe.
// M padded 4->16 with a zero multiplier (no EXEC divergence around WMMA).
__global__ void __launch_bounds__(128) proj_wmma_kernel(
    const float* __restrict__ h, const float* __restrict__ W,
    const float* __restrict__ bias, float* __restrict__ proj,
    float* __restrict__ partials) {
  const int wave = blockIdx.x * 4 + (threadIdx.x >> 5);   // 0..255
  const int lane = threadIdx.x & 31;
  const int n0 = wave * 16;                               // column tile base
  const int mrow = lane & 15;                             // A: M row / B: N col
  const int half = lane >> 4;                             // K split: 0 -> K{0,1}, 1 -> K{2,3}
  const float amask = (mrow < 4) ? 1.f : 0.f;             // pad M rows 4..15 with zeros

  // A fragment source: h[mrow][k + half*2 + {0,1}] (consecutive K -> float2 load)
  const float* hrow = h + (size_t)(mrow & 3) * DF + half * 2;
  // B fragment source: W[n0+mrow][k + half*2 + {0,1}]
  const float* wrow = W + (size_t)(n0 + mrow) * DF + half * 2;

  v8f acc = {};
  for (int k = 0; k < DF; k += 4) {
    v2f a = *(const v2f*)(hrow + k);
    a.x *= amask;
    a.y *= amask;
    v2f bf = *(const v2f*)(wrow + k);
    acc = __builtin_amdgcn_wmma_f32_16x16x4_f32(
        /*neg_a=*/false, a, /*neg_b=*/false, bf,
        /*c_mod=*/(short)0, acc, /*reuse_a=*/false, /*reuse_b=*/false);
  }

  // D layout: VGPR r, lanes 0-15 -> (M=r, N=lane). Valid rows M=0..3 live in acc[0..3].
  const float bn = bias[n0 + mrow];
#pragma unroll
  for (int m = 0; m < 4; ++m) {
    float v = acc[m] + bn;
    float sq = (half == 0) ? v * v : 0.f;
#pragma unroll
    for (int off = 16; off >= 1; off >>= 1) sq += __shfl_xor(sq, off, 32);
    if (lane == 0) partials[m * 256 + wave] = sq;           // deterministic partials
    if (half == 0) proj[m * DM + n0 + mrow] = v;
  }
}

// Kernel 3: reduce 256 partials per row -> scale[b] = magnitude / max(||proj||, eps)
__global__ void __launch_bounds__(256) finalize_kernel(
    const float* __restrict__ partials, const float* __restrict__ magnitude,
    float* __restrict__ scale) {
  __shared__ float sdata[32];
  const int tid = threadIdx.x;
#pragma unroll
  for (int m = 0; m < 4; ++m) {
    float tot = block_reduce_sum(partials[m * 256 + tid], sdata);
    if (tid == 0) scale[m] = magnitude[m] / fmaxf(sqrtf(tot), NORM_EPS_);
  }
}

// Kernel 4: streaming copy out = x (b128 non-temporal), fusing the delta add
// into the 4 grafted rows (block-uniform branch). This is the 512 MB / 23.3 TB/s
// bandwidth-bound part (~22 us floor).
__global__ void __launch_bounds__(256) copy_graft_kernel(
    const v4f* __restrict__ x, v4f* __restrict__ out,
    const float* __restrict__ proj, const float* __restrict__ scale,
    const int* __restrict__ last_idx) {
  const int row = blockIdx.x;          // 0 .. B*S-1
  const int b = row >> 12;             // S == 4096
  const int s = row & (S_ - 1);
  const bool graft = (s == last_idx[b]);
  const float sc = graft ? scale[b] : 0.f;
  const v4f* prow = (const v4f*)(proj + (size_t)b * DM);
  const size_t base = (size_t)row * (DM / 4);
#pragma unroll
  for (int i = 0; i < 4; ++i) {
    const int e = threadIdx.x + i * 256;
    v4f v = __builtin_nontemporal_load(&x[base + e]);
    if (graft) {
      v4f p = prow[e];
      v += p * sc;
    }
    __builtin_nontemporal_store(v, &out[base + e]);
  }
}

extern "C" void kernel_launch(void* const* d_in, const int* in_sizes, int n_in,
                              void* d_out, int out_size, void* d_ws, size_t ws_size,
                              hipStream_t stream) {
  (void)in_sizes; (void)n_in; (void)out_size; (void)ws_size;
  const float* x     = (const float*)d_in[0];   // [4,4096,4096] f32
  const float* fac   = (const float*)d_in[1];   // [4,512] f32
  /* d_in[2] token_ids (int64) unused by reference math */
  const int*   mask  = (const int*)d_in[3];     // [4,4096] i32
  const float* gamma = (const float*)d_in[4];   // [512]
  const float* beta  = (const float*)d_in[5];   // [512]
  const float* W     = (const float*)d_in[6];   // [4096,512]
  const float* bias  = (const float*)d_in[7];   // [4096]
  float* out = (float*)d_out;

  // Workspace layout (floats): ~78 KB total
  float* ws        = (float*)d_ws;
  float* h         = ws;                     // 4*512  = 2048
  float* proj      = ws + 2048;              // 4*4096 = 16384
  float* partials  = ws + 2048 + 16384;      // 4*256  = 1024
  float* magnitude = partials + 1024;        // 4
  float* scale     = magnitude + 4;          // 4
  int*   last_idx  = (int*)(scale + 4);      // 4

  prep_kernel<<<B_, 256, 0, stream>>>(x, fac, mask, gamma, beta, h, magnitude, last_idx);
  proj_wmma_kernel<<<64, 128, 0, stream>>>(h, W, bias, proj, partials);
  finalize_kernel<<<1, 256, 0, stream>>>(partials, magnitude, scale);
  copy_graft_kernel<<<B_ * S_, 256, 0, stream>>>((const v4f*)x, (v4f*)out, proj, scale, last_idx);
}